// Model_91036126806515
// MI455X (gfx1250) — compile-verified
//
#include <hip/hip_runtime.h>
#include <cstdint>
#include <cstddef>

// Jagged layer-norm, MI455X (gfx1250).
// Assumptions from the reference: M == 64 columns (16 float4 quads per row),
// f32 in/out, B = n_offsets - 1 segments (B <= MAX_SEGS).

#define EPSILON 1e-6f

typedef float v4f __attribute__((ext_vector_type(4)));  // native vector for NT store

constexpr int M_COLS   = 64;
constexpr int TB       = 256;            // threads per block (8 wave32)
constexpr int WAVES    = TB / 32;
constexpr int UNROLL   = 4;              // quads per lane per staged tile
constexpr int WTILE    = UNROLL * 32;    // quads per wave-tile (128 quads = 2 KB)
constexpr int TILE_Q   = TB * UNROLL;    // quads per block-tile (1024)
constexpr int MAX_SEGS = 64;

// --- segment lookup: largest i with off[i] <= row (searchsorted-right - 1) ---
__device__ __forceinline__ int find_seg(int row, const int* s_off, int B) {
    int lo = 0, hi = B;
    while (hi - lo > 1) {
        int mid = (lo + hi) >> 1;
        if (row >= s_off[mid]) lo = mid; else hi = mid;
    }
    return lo;
}

// Issue UNROLL async B128 copies: global -> this wave's LDS staging buffer.
// GVS mode: 64-bit SGPR base + signed 32-bit VGPR byte offset (<256MB, fits).
__device__ __forceinline__ void issue_tile_async(const float* base,
                                                 uint32_t ldsLaneAddr,
                                                 size_t qbase, int lane) {
#pragma unroll
    for (int i = 0; i < UNROLL; ++i) {
        uint32_t byteOff = (uint32_t)((qbase + (size_t)(i * 32) + (size_t)lane) * 16u);
        uint32_t ldsAddr = ldsLaneAddr + (uint32_t)(i * 32 * 16);
        asm volatile("global_load_async_to_lds_b128 %0, %1, %2"
                     :
                     : "v"(ldsAddr), "v"(byteOff), "s"(base)
                     : "memory");
    }
}

__global__ void k_zero(float* __restrict__ ws, int B) {
    int t = threadIdx.x;
    if (t < 2 * B) ws[t] = 0.0f;
}

// Pass 1: per-segment sum and sum-of-squares via async-to-LDS staging.
__global__ void k_reduce(const float* __restrict__ values,
                         const int* __restrict__ offsets,
                         float* __restrict__ ws,
                         int B, size_t totalQuads, size_t numFullTiles) {
    __shared__ int    s_off[MAX_SEGS + 1];
    __shared__ float  s_sum[MAX_SEGS];
    __shared__ float  s_sq [MAX_SEGS];
    __shared__ float4 stage[WAVES][2][WTILE];   // ping-pong per wave: 32 KB

    const int tid  = threadIdx.x;
    const int wave = tid >> 5;
    const int lane = tid & 31;

    if (tid < B)  { s_sum[tid] = 0.0f; s_sq[tid] = 0.0f; }
    if (tid <= B) { s_off[tid] = offsets[tid]; }
    __syncthreads();

    const uint32_t ldsBuf0 = (uint32_t)(uintptr_t)(&stage[wave][0][lane]);
    const uint32_t ldsBuf1 = (uint32_t)(uintptr_t)(&stage[wave][1][lane]);
    const size_t   waveOff = (size_t)wave * WTILE;
    const size_t   gstride = (size_t)gridDim.x;

    size_t tile = blockIdx.x;
    int    buf  = 0;
    if (tile < numFullTiles)
        issue_tile_async(values, ldsBuf0, tile * TILE_Q + waveOff, lane);

    for (; tile < numFullTiles; tile += gstride) {
        const size_t nextTile = tile + gstride;
        if (nextTile < numFullTiles) {
            issue_tile_async(values, buf ? ldsBuf0 : ldsBuf1,
                             nextTile * TILE_Q + waveOff, lane);
            // UNROLL copies still in flight for the next tile; ours are done.
            asm volatile("s_wait_asynccnt 4" ::: "memory");
        } else {
            asm volatile("s_wait_asynccnt 0" ::: "memory");
        }

        const size_t qbase = tile * TILE_Q + waveOff;
        const float4* wst  = stage[wave][buf];
#pragma unroll
        for (int i = 0; i < UNROLL; ++i) {
            float4 v  = wst[i * 32 + lane];
            float  s  = (v.x + v.y) + (v.z + v.w);
            float  sq = v.x * v.x + v.y * v.y + v.z * v.z + v.w * v.w;
            // 16 lanes cover one full row (16 quads * 4 floats = 64 cols)
#pragma unroll
            for (int m = 1; m < 16; m <<= 1) {
                s  += __shfl_xor(s,  m);
                sq += __shfl_xor(sq, m);
            }
            if ((lane & 15) == 0) {
                int row = (int)((qbase + (size_t)(i * 32) + (size_t)lane) >> 4);
                int seg = find_seg(row, s_off, B);
                atomicAdd(&s_sum[seg], s);
                atomicAdd(&s_sq [seg], sq);
            }
        }
        buf ^= 1;
    }

    // Tail (quads not covered by full tiles): direct loads, rare path.
    const size_t tailStart = numFullTiles * TILE_Q;
    for (size_t q = tailStart + (size_t)blockIdx.x * TB + tid;
         q < totalQuads; q += gstride * TB) {
        float4 v  = ((const float4*)values)[q];
        float  s  = (v.x + v.y) + (v.z + v.w);
        float  sq = v.x * v.x + v.y * v.y + v.z * v.z + v.w * v.w;
        int row = (int)(q >> 4);
        int seg = find_seg(row, s_off, B);
        atomicAdd(&s_sum[seg], s);
        atomicAdd(&s_sq [seg], sq);
    }

    __syncthreads();
    if (tid < B) {
        atomicAdd(&ws[tid],     s_sum[tid]);
        atomicAdd(&ws[B + tid], s_sq [tid]);
    }
}

// Pass 2: normalize. Reverse tile order so the tail of `values` (still resident
// in the 192 MB L2 after pass 1's ascending read) is reused before eviction.
__global__ void k_norm(const float* __restrict__ values,
                       const int* __restrict__ offsets,
                       const float* __restrict__ ws,
                       float* __restrict__ out,
                       int B, size_t totalQuads) {
    __shared__ int   s_off[MAX_SEGS + 1];
    __shared__ float s_mean[MAX_SEGS];
    __shared__ float s_inv [MAX_SEGS];

    const int tid = threadIdx.x;
    if (tid <= B) s_off[tid] = offsets[tid];
    if (tid < B) {
        float len   = (float)(offsets[tid + 1] - offsets[tid]);
        float denom = fmaxf(len * (float)M_COLS, 1.0f);
        float mean  = ws[tid] / denom;
        float var   = ws[B + tid] / denom - mean * mean;
        var         = fmaxf(var, 0.0f);
        s_mean[tid] = mean;
        s_inv [tid] = 1.0f / sqrtf(var + EPSILON);
    }
    __syncthreads();

    const v4f* in4  = (const v4f*)values;
    v4f*       out4 = (v4f*)out;
    const long long numTiles = (long long)((totalQuads + TB - 1) / TB);

    for (long long t = numTiles - 1 - (long long)blockIdx.x; t >= 0;
         t -= (long long)gridDim.x) {
        size_t q = (size_t)t * TB + (size_t)tid;
        if (q < totalQuads) {
            v4f    v   = in4[q];
            int    row = (int)(q >> 4);
            int    seg = find_seg(row, s_off, B);
            float  m   = s_mean[seg];
            float  iv  = s_inv [seg];
            v4f    r   = (v - m) * iv;                 // elementwise on ext-vector
            __builtin_nontemporal_store(r, &out4[q]);  // output never re-read
        }
    }
}

extern "C" void kernel_launch(void* const* d_in, const int* in_sizes, int n_in,
                              void* d_out, int out_size, void* d_ws, size_t ws_size,
                              hipStream_t stream) {
    const float* values  = (const float*)d_in[0];
    const int*   offsets = (const int*)d_in[1];
    float*       out     = (float*)d_out;
    float*       ws      = (float*)d_ws;

    const int    B            = in_sizes[1] - 1;             // 16 segments
    const size_t totalElems   = (size_t)in_sizes[0];         // total * M
    const size_t totalQuads   = totalElems / 4;              // 16 M float4s
    const size_t numFullTiles = totalQuads / TILE_Q;

    const int grid = 2048;   // grid-stride loops handle any size

    k_zero  <<<1,    TB, 0, stream>>>(ws, B);
    k_reduce<<<grid, TB, 0, stream>>>(values, offsets, ws, B, totalQuads, numFullTiles);
    k_norm  <<<grid, TB, 0, stream>>>(values, offsets, ws, out, B, totalQuads);
}